// SSALayer_42348377538687
// MI455X (gfx1250) — compile-verified
//
#include <hip/hip_runtime.h>
#include <hip/hip_bf16.h>

typedef __attribute__((ext_vector_type(2))) float v2f;
typedef __attribute__((ext_vector_type(8))) float v8f;

#define Bb 32
#define Cc 64
#define Nn 16384   // H*W = 128*128
#define Mm 8

// ---------------- k0: zero the global accumulators ----------------
__global__ void k0_init(float* stat, float* unorm) {
    int t = threadIdx.x;
    if (t < 32) stat[t] = 0.0f;
    if (t < 64) unorm[t] = 0.0f;
}

// ---------------- k1: conv1x1 (f,g,h) via f32 WMMA + BN stats ----------------
// One wave = one 16-pixel tile. Chain FG: B cols 0-7 = f_w, 8-15 = g_w.
// Chain H: B cols 0-7 = h_w (cols 8-15 dummy, discarded).
__global__ __launch_bounds__(256) void k1_conv_wmma(
    const float* __restrict__ x,
    const float* __restrict__ f_w, const float* __restrict__ f_b,
    const float* __restrict__ g_w, const float* __restrict__ g_b,
    const float* __restrict__ h_w, const float* __restrict__ h_b,
    float* __restrict__ f_ws, float* __restrict__ g_ws, float* __restrict__ h_ws,
    float* __restrict__ stat)
{
    __shared__ float ssum[16];
    __shared__ float s2sum[16];
    int tid = threadIdx.x;
    int lane = tid & 31;
    int wave = tid >> 5;
    if (tid < 16) { ssum[tid] = 0.0f; s2sum[tid] = 0.0f; }
    __syncthreads();

    int tile = blockIdx.x * 8 + wave;        // 0..32767
    int b    = tile >> 10;                   // tile / 1024
    int n0   = (tile & 1023) << 4;           // *16 pixels

    int j    = lane & 15;                    // A row (pixel) / B,D col (out channel)
    int koff = (lane >> 4) << 1;             // 0 or 2

    const float* xb  = x + (size_t)b * Cc * Nn + (size_t)(n0 + j);
    const float* wfg = (j < 8 ? f_w : g_w) + (size_t)(j & 7) * Cc + koff;
    const float* wh  = (j < 8 ? h_w : f_w) + (size_t)(j & 7) * Cc + koff; // dummy for j>=8

    v8f accFG = {};
    v8f accH  = {};
#pragma unroll
    for (int kk = 0; kk < 16; ++kk) {
        int c = 4 * kk + koff;
        v2f a, bf, bh;
        a.x  = xb[(size_t)c * Nn];
        a.y  = xb[(size_t)(c + 1) * Nn];
        bf.x = wfg[4 * kk];
        bf.y = wfg[4 * kk + 1];
        bh.x = wh[4 * kk];
        bh.y = wh[4 * kk + 1];
        accFG = __builtin_amdgcn_wmma_f32_16x16x4_f32(false, a, false, bf, (short)0, accFG, false, false);
        accH  = __builtin_amdgcn_wmma_f32_16x16x4_f32(false, a, false, bh, (short)0, accH, false, false);
    }

    // D layout: vgpr r, lane L -> pixel n0 + r + 8*(L>=16), channel j = L%16
    float biasFG = (j < 8) ? f_b[j] : g_b[j - 8];
    float biasH  = (j < 8) ? h_b[j] : 0.0f;
    size_t rowbase = ((size_t)b * Nn + (size_t)(n0 + 8 * (lane >> 4))) * Mm + (j & 7);
    float* outFG = (j < 8 ? f_ws : g_ws) + rowbase;

    float s = 0.0f, s2 = 0.0f;
#pragma unroll
    for (int r = 0; r < 8; ++r) {
        float v = accFG[r] + biasFG;
        outFG[(size_t)r * Mm] = v;
        s += v; s2 += v * v;
    }
    if (j < 8) {
        float* outH = h_ws + rowbase;
#pragma unroll
        for (int r = 0; r < 8; ++r) outH[(size_t)r * Mm] = accH[r] + biasH;
    }

    // combine the two half-wave K partners (same channel, other 8 pixels)
    s  += __shfl_xor(s, 16, 32);
    s2 += __shfl_xor(s2, 16, 32);
    if (lane < 16) { atomicAdd(&ssum[j], s); atomicAdd(&s2sum[j], s2); }
    __syncthreads();
    if (tid < 16) {
        int base = (tid < 8) ? 0 : 16;
        int idx  = tid & 7;
        atomicAdd(stat + base + idx,     ssum[tid]);
        atomicAdd(stat + base + 8 + idx, s2sum[tid]);
    }
}

// ---------------- k2: BN finalize + per-batch Gram (F^TF, G^TG) via WMMA ----------------
// Combined [N,16] matrix (f cols 0-7, g cols 8-15) used as both A and B:
// accumulator diag blocks = F^TF and G^TG.
__global__ __launch_bounds__(256) void k2_gram_wmma(
    const float* __restrict__ f_ws, const float* __restrict__ g_ws,
    const float* __restrict__ stat,
    const float* __restrict__ f_gamma, const float* __restrict__ f_beta,
    const float* __restrict__ g_gamma, const float* __restrict__ g_beta,
    float* __restrict__ ws_comb /*[B][16*16]*/, float* __restrict__ bncoef /*[32]*/)
{
    __shared__ float red[256];
    int tid = threadIdx.x, lane = tid & 31, wave = tid >> 5;
    int b = blockIdx.x;
    int j = lane & 15;
    int koff = (lane >> 4) << 1;

    const float cnt = (float)Bb * (float)Nn;
    float sum  = stat[(j < 8 ? 0 : 16) + (j & 7)];
    float sq   = stat[(j < 8 ? 8 : 24) + (j & 7)];
    float mean = sum / cnt;
    float var  = sq / cnt - mean * mean;
    float gam  = (j < 8) ? f_gamma[j] : g_gamma[j - 8];
    float bet  = (j < 8) ? f_beta[j]  : g_beta[j - 8];
    float aco  = gam * rsqrtf(var + 1e-5f);
    float cco  = bet - aco * mean;

    const float* basep = (j < 8 ? f_ws : g_ws) + (size_t)b * Nn * Mm + (j & 7);

    v8f acc = {};
    for (int i = 0; i < 512; ++i) {            // 4096 pixel-quads / 8 waves
        int n0 = (wave + (i << 3)) << 2;       // quad * 4
        size_t off = (size_t)(n0 + koff) * Mm;
        float r0 = basep[off];
        float r1 = basep[off + Mm];
        v2f a;
        a.x = fmaxf(aco * r0 + cco, 0.0f);
        a.y = fmaxf(aco * r1 + cco, 0.0f);
        acc = __builtin_amdgcn_wmma_f32_16x16x4_f32(false, a, false, a, (short)0, acc, false, false);
    }

    red[tid] = 0.0f;
    __syncthreads();
#pragma unroll
    for (int r = 0; r < 8; ++r) atomicAdd(&red[r * 32 + lane], acc[r]);
    __syncthreads();

    // write canonical row-major 16x16: slot(row,col) = (row%8)*32 + col + 16*(row/8)
    int rowE = tid >> 4, colE = tid & 15;
    ws_comb[b * 256 + tid] = red[(rowE & 7) * 32 + colE + ((rowE >> 3) << 4)];

    if (b == 0 && tid < 16) {  // publish BN coefs once: af[8] cf[8] ag[8] cg[8]
        int base = (tid < 8) ? 0 : 16;
        bncoef[base + (tid & 7)]     = aco;
        bncoef[base + 8 + (tid & 7)] = cco;
    }
}

// ---------------- k3: 64 independent 8x8 power iterations ----------------
__global__ void k3_power(const float* __restrict__ ws_comb,
                         const float* __restrict__ v0f, const float* __restrict__ v0g,
                         float* __restrict__ vfg /*[2][B][8]*/)
{
    int t = threadIdx.x;
    if (t >= 64) return;
    int b = t >> 1, br = t & 1, off = br * 8;
    float W[8][8];
    for (int i = 0; i < 8; ++i)
        for (int k = 0; k < 8; ++k)
            W[i][k] = ws_comb[b * 256 + (i + off) * 16 + (k + off)];
    const float* v0p = (br ? v0g : v0f) + b * 8;
    float v[8];
    for (int i = 0; i < 8; ++i) v[i] = v0p[i];
    for (int it = 0; it < 30; ++it) {
        float nv[8], nrm = 0.0f;
        for (int i = 0; i < 8; ++i) {
            float s = 0.0f;
            for (int k = 0; k < 8; ++k) s += W[i][k] * v[k];
            nv[i] = s; nrm += s * s;
        }
        float inv = rsqrtf(nrm);
        for (int i = 0; i < 8; ++i) v[i] = nv[i] * inv;
    }
    float* o = vfg + (size_t)(br * Bb + b) * 8;
    for (int i = 0; i < 8; ++i) o[i] = v[i];
}

// ---------------- k4a: u = BN(raw) @ v per pixel + per-batch ||u||^2 ----------------
__global__ __launch_bounds__(256) void k4a_u(
    const float* __restrict__ f_ws, const float* __restrict__ g_ws,
    const float* __restrict__ bncoef, const float* __restrict__ vfg,
    float* __restrict__ ufraw, float* __restrict__ ugraw, float* __restrict__ unorm /*[2][B]*/)
{
    __shared__ float sf, sg;
    int tid = threadIdx.x;
    int b = blockIdx.x >> 6;                    // 64 blocks per batch
    int n = ((blockIdx.x & 63) << 8) + tid;
    if (tid == 0) { sf = 0.0f; sg = 0.0f; }
    __syncthreads();

    float uf = 0.0f, ug = 0.0f;
    const float* fr = f_ws + ((size_t)b * Nn + n) * Mm;
    const float* gr = g_ws + ((size_t)b * Nn + n) * Mm;
#pragma unroll
    for (int m = 0; m < 8; ++m) {
        float af = bncoef[m],      cf = bncoef[8 + m];
        float ag = bncoef[16 + m], cg = bncoef[24 + m];
        uf += fmaxf(af * fr[m] + cf, 0.0f) * vfg[b * 8 + m];
        ug += fmaxf(ag * gr[m] + cg, 0.0f) * vfg[(Bb + b) * 8 + m];
    }
    ufraw[(size_t)b * Nn + n] = uf;
    ugraw[(size_t)b * Nn + n] = ug;
    atomicAdd(&sf, uf * uf);
    atomicAdd(&sg, ug * ug);
    __syncthreads();
    if (tid == 0) { atomicAdd(&unorm[b], sf); atomicAdd(&unorm[Bb + b], sg); }
}

// ---------------- k4b: coef = sum_n ug*h ; fold v_w + normalizations ----------------
__global__ __launch_bounds__(256) void k4b_coef(
    const float* __restrict__ h_ws, const float* __restrict__ ugraw,
    const float* __restrict__ unorm, const float* __restrict__ v_w,
    float* __restrict__ w2 /*[B][64]*/)
{
    __shared__ float cs[8];
    int tid = threadIdx.x, b = blockIdx.x;
    if (tid < 8) cs[tid] = 0.0f;
    __syncthreads();
    float acc[8] = {};
    const float* hb = h_ws + (size_t)b * Nn * Mm;
    const float* ub = ugraw + (size_t)b * Nn;
    for (int n = tid; n < Nn; n += 256) {
        float u = ub[n];
        const float* hr = hb + (size_t)n * Mm;
#pragma unroll
        for (int m = 0; m < 8; ++m) acc[m] += u * hr[m];
    }
#pragma unroll
    for (int m = 0; m < 8; ++m) atomicAdd(&cs[m], acc[m]);
    __syncthreads();
    if (tid < 64) {
        float invng = rsqrtf(unorm[Bb + b]);
        float invnf = rsqrtf(unorm[b]);
        float s = 0.0f;
        for (int m = 0; m < 8; ++m) s += v_w[tid * 8 + m] * cs[m];
        w2[b * 64 + tid] = s * invng * invnf;
    }
}

// ---------------- k5: out = x + uf[n]*w2[b,c] + v_b[c]  (float4 stream) ----------------
__global__ __launch_bounds__(256) void k5_out(
    const float* __restrict__ x, const float* __restrict__ ufraw,
    const float* __restrict__ w2, const float* __restrict__ v_b,
    float* __restrict__ out)
{
    size_t q = (size_t)blockIdx.x * 256 + threadIdx.x;   // quad index, total 8388608
    int b   = (int)(q >> 18);            // 262144 quads per batch
    int rem = (int)(q & 262143u);
    int c   = rem >> 12;                 // 4096 quads per channel
    int nq  = rem & 4095;
    float sc = w2[b * 64 + c];
    float vb = v_b[c];
    float4 xv = ((const float4*)x)[q];
    float4 uv = ((const float4*)(ufraw + (size_t)b * Nn))[nq];
    float4 o;
    o.x = xv.x + uv.x * sc + vb;
    o.y = xv.y + uv.y * sc + vb;
    o.z = xv.z + uv.z * sc + vb;
    o.w = xv.w + uv.w * sc + vb;
    ((float4*)out)[q] = o;
}

extern "C" void kernel_launch(void* const* d_in, const int* in_sizes, int n_in,
                              void* d_out, int out_size, void* d_ws, size_t ws_size,
                              hipStream_t stream) {
    const float* x       = (const float*)d_in[0];
    const float* f_w     = (const float*)d_in[1];
    const float* f_b     = (const float*)d_in[2];
    const float* f_gamma = (const float*)d_in[3];
    const float* f_beta  = (const float*)d_in[4];
    const float* g_w     = (const float*)d_in[5];
    const float* g_b     = (const float*)d_in[6];
    const float* g_gamma = (const float*)d_in[7];
    const float* g_beta  = (const float*)d_in[8];
    const float* h_w     = (const float*)d_in[9];
    const float* h_b     = (const float*)d_in[10];
    const float* v_w     = (const float*)d_in[11];
    const float* v_b     = (const float*)d_in[12];
    const float* v0f     = (const float*)d_in[13];
    const float* v0g     = (const float*)d_in[14];
    float* out = (float*)d_out;

    // workspace layout (floats)
    float* ws = (float*)d_ws;
    const size_t SZ = (size_t)Bb * Nn * Mm;    // 4,194,304
    float* f_ws    = ws;                 ws += SZ;
    float* g_ws    = ws;                 ws += SZ;
    float* h_ws    = ws;                 ws += SZ;
    float* ufraw   = ws;                 ws += (size_t)Bb * Nn;
    float* ugraw   = ws;                 ws += (size_t)Bb * Nn;
    float* ws_comb = ws;                 ws += Bb * 256;
    float* vfg     = ws;                 ws += 2 * Bb * 8;
    float* stat    = ws;                 ws += 32;
    float* unorm   = ws;                 ws += 64;
    float* bncoef  = ws;                 ws += 32;
    float* w2      = ws;                 ws += Bb * 64;

    k0_init<<<1, 64, 0, stream>>>(stat, unorm);
    k1_conv_wmma<<<4096, 256, 0, stream>>>(x, f_w, f_b, g_w, g_b, h_w, h_b,
                                           f_ws, g_ws, h_ws, stat);
    k2_gram_wmma<<<Bb, 256, 0, stream>>>(f_ws, g_ws, stat,
                                         f_gamma, f_beta, g_gamma, g_beta,
                                         ws_comb, bncoef);
    k3_power<<<1, 64, 0, stream>>>(ws_comb, v0f, v0g, vfg);
    k4a_u<<<Bb * 64, 256, 0, stream>>>(f_ws, g_ws, bncoef, vfg, ufraw, ugraw, unorm);
    k4b_coef<<<Bb, 256, 0, stream>>>(h_ws, ugraw, unorm, v_w, w2);
    k5_out<<<32768, 256, 0, stream>>>(x, ufraw, w2, v_b, out);
}